// ActionRecognitionCRF_69123203662574
// MI455X (gfx1250) — compile-verified
//
#include <hip/hip_runtime.h>

#define T_FRAMES 16384
#define INPUT_DIM 4096
#define PSZ 64
#define START_TAG 62
#define STOP_TAG 63
#define NEGV (-10000.0f)
#define CHUNK 128
#define NCHUNK (T_FRAMES / CHUNK)

typedef float v2f __attribute__((ext_vector_type(2)));
typedef float v8f __attribute__((ext_vector_type(8)));

// ---------------------------------------------------------------------------
// Kernel 1: feats = relu(video @ W^T + b) using V_WMMA_F32_16X16X4_F32.
// Grid: T/16 blocks x 128 threads (4 waves); each wave computes one 16x16 tile.
// A fragment: lane l<16 holds row M=l, K = k0+0 / k0+1 (VGPR0/1);
//             lane l>=16 holds row M=l-16, K = k0+2 / k0+3.  (ISA 7.12.2)
// B fragment mirrors the half-wave K split with N = lane&15.
// Both map to per-lane contiguous b64 loads of row-major video / W.
// ---------------------------------------------------------------------------
__global__ __launch_bounds__(128) void emis_kernel(const float* __restrict__ video,
                                                   const float* __restrict__ W,
                                                   const float* __restrict__ bias,
                                                   float* __restrict__ feats) {
    const int lane = threadIdx.x & 31;
    const int wave = threadIdx.x >> 5;      // 0..3 -> n0 = wave*16
    const int row0 = blockIdx.x * 16;
    const int n0   = wave * 16;
    const int half = lane >> 4;             // which 16-lane half
    const int l16  = lane & 15;

    const float* aptr = video + (size_t)(row0 + l16) * INPUT_DIM + half * 2;
    const float* bptr = W     + (size_t)(n0   + l16) * INPUT_DIM + half * 2;

    v8f acc = {};
#pragma unroll 8
    for (int k = 0; k < INPUT_DIM; k += 4) {
        v2f a = *(const v2f*)(aptr + k);
        v2f b = *(const v2f*)(bptr + k);
        // 8 args: (neg_a, A, neg_b, B, c_mod, C, reuse_a, reuse_b)
        acc = __builtin_amdgcn_wmma_f32_16x16x4_f32(false, a, false, b,
                                                    (short)0, acc, false, false);
    }

    // C/D layout: VGPR r, lanes 0-15: (M=r, N=lane); lanes 16-31: (M=8+r, N=lane-16)
    float bv = bias[n0 + l16];
#pragma unroll
    for (int r = 0; r < 8; ++r) {
        int m = r + half * 8;
        float v = acc[r] + bv;
        v = v > 0.0f ? v : 0.0f;
        feats[(size_t)(row0 + m) * PSZ + n0 + l16] = v;
    }
}

// ---------------------------------------------------------------------------
// Kernel 2: sequential Viterbi forward. 1 block x 64 threads (2 waves),
// lane = next-state. Transition row in VGPRs; fv and the chunk-composed
// backpointer map G double-buffered in LDS -> one barrier per step.
// Emits: bptrs (byte per state/step), comp (chunk-composed maps), final fv.
// ---------------------------------------------------------------------------
__global__ __launch_bounds__(64) void viterbi_fwd(const float* __restrict__ feats,
                                                  const float* __restrict__ trans,
                                                  unsigned char* __restrict__ bptrs,
                                                  unsigned char* __restrict__ comp,
                                                  float* __restrict__ fvfinal) {
    __shared__ float fvbuf[2][PSZ];
    __shared__ int   gbuf[2][PSZ];
    const int tid = threadIdx.x;            // next-state

    float4 tr[16];
#pragma unroll
    for (int j = 0; j < 16; ++j)
        tr[j] = *(const float4*)(trans + (size_t)tid * PSZ + j * 4);

    fvbuf[0][tid] = (tid == START_TAG) ? 0.0f : NEGV;
    gbuf[0][tid]  = tid;                    // identity map for first chunk
    __syncthreads();

    int p = 0;
    float fcur = feats[tid];                // emission for t=0
    float fvlast = 0.0f;
    for (int t = 0; t < T_FRAMES; ++t) {
        float fnext = (t + 1 < T_FRAMES) ? feats[(size_t)(t + 1) * PSZ + tid] : 0.0f;

        float m = -3.0e38f;
        int b = 0;
#pragma unroll
        for (int j = 0; j < 16; ++j) {
            float4 fv4 = *(const float4*)&fvbuf[p][j * 4];
            float s;
            s = fv4.x + tr[j].x; if (s > m) { m = s; b = j * 4 + 0; }
            s = fv4.y + tr[j].y; if (s > m) { m = s; b = j * 4 + 1; }
            s = fv4.z + tr[j].z; if (s > m) { m = s; b = j * 4 + 2; }
            s = fv4.w + tr[j].w; if (s > m) { m = s; b = j * 4 + 3; }
        }
        float fvnew = m + fcur;
        int gnew = gbuf[p][b];              // compose: G_new(x) = G_prev[bptr_t[x]]
        bptrs[(size_t)t * PSZ + tid] = (unsigned char)b;

        int gstore = gnew;
        if ((t & (CHUNK - 1)) == (CHUNK - 1)) {     // flush chunk map, reset to id
            comp[(size_t)(t >> 7) * PSZ + tid] = (unsigned char)gnew;
            gstore = tid;
        }
        fvbuf[p ^ 1][tid] = fvnew;
        gbuf[p ^ 1][tid]  = gstore;
        __syncthreads();                    // publish buf p^1; reads of p are done
        p ^= 1;
        fcur = fnext;
        fvlast = fvnew;
    }
    fvfinal[tid] = fvlast;
}

// ---------------------------------------------------------------------------
// Kernel 3: terminal argmax (first-index, matching jnp.argmax) + chunk-level
// backtrace through the composed maps (only NCHUNK dependent lookups).
// ---------------------------------------------------------------------------
__global__ void viterbi_term(const float* __restrict__ fvfinal,
                             const float* __restrict__ trans,
                             const unsigned char* __restrict__ comp,
                             int* __restrict__ boundary,
                             float* __restrict__ out) {
    if (threadIdx.x == 0) {
        float best = -3.0e38f; int bi = 0;
        for (int x = 0; x < PSZ; ++x) {
            float s = fvfinal[x] + trans[STOP_TAG * PSZ + x];
            if (s > best) { best = s; bi = x; }
        }
        out[0] = best;                      // path_score
        int e = bi;                         // tag entering topmost chunk
        for (int c = NCHUNK - 1; c >= 0; --c) {
            boundary[c] = e;
            e = comp[(size_t)c * PSZ + e];
        }
    }
}

// ---------------------------------------------------------------------------
// Kernel 4: per-chunk backtrace in parallel. Lane c walks chunk c's 128 raw
// backpointers starting from its boundary tag. out[1+t] = best_path[t].
// ---------------------------------------------------------------------------
__global__ __launch_bounds__(NCHUNK) void backtrace_kernel(const unsigned char* __restrict__ bptrs,
                                                           const int* __restrict__ boundary,
                                                           float* __restrict__ out) {
    int c = threadIdx.x;                    // chunk id
    int carry = boundary[c];
    for (int i = CHUNK - 1; i >= 0; --i) {
        int t = c * CHUNK + i;
        out[1 + t] = (float)carry;          // best_path[t] = carry_{t+1}
        carry = bptrs[(size_t)t * PSZ + carry];
    }
}

// ---------------------------------------------------------------------------
extern "C" void kernel_launch(void* const* d_in, const int* in_sizes, int n_in,
                              void* d_out, int out_size, void* d_ws, size_t ws_size,
                              hipStream_t stream) {
    const float* video = (const float*)d_in[0];   // [16384, 4096]
    const float* W     = (const float*)d_in[1];   // [64, 4096]
    const float* bias  = (const float*)d_in[2];   // [64]
    const float* trans = (const float*)d_in[3];   // [64, 64]
    float* out = (float*)d_out;                   // [1 + 16384]

    char* ws = (char*)d_ws;
    float*         feats   = (float*)(ws);                       // 4 MB
    unsigned char* bptrs   = (unsigned char*)(ws + 4194304);     // 1 MB
    unsigned char* comp    = (unsigned char*)(ws + 5242880);     // 8 KB
    float*         fvfinal = (float*)(ws + 5251072);             // 256 B
    int*           boundary= (int*)(ws + 5251328);               // 512 B

    emis_kernel<<<T_FRAMES / 16, 128, 0, stream>>>(video, W, bias, feats);
    viterbi_fwd<<<1, PSZ, 0, stream>>>(feats, trans, bptrs, comp, fvfinal);
    viterbi_term<<<1, 64, 0, stream>>>(fvfinal, trans, comp, boundary, out);
    backtrace_kernel<<<1, NCHUNK, 0, stream>>>(bptrs, boundary, out);
}